// GroupGeM_25666724561287
// MI455X (gfx1250) — compile-verified
//
#include <hip/hip_runtime.h>

typedef float v2f __attribute__((ext_vector_type(2)));
typedef float v4f __attribute__((ext_vector_type(4)));
typedef float v8f __attribute__((ext_vector_type(8)));

#define GEM_EPS 1e-6f

// Hardware transcendentals (v_exp_f32 / v_log_f32), no header dependency.
__device__ __forceinline__ float fast_exp2(float x) { return __builtin_amdgcn_exp2f(x); }
__device__ __forceinline__ float fast_log2(float x) { return __builtin_amdgcn_logf(x); }

// One wave32 handles 16 consecutive (batch*channel) rows.
// Lane l (<16)  : streams floats [0, hw/2)  of row (bc0 + l)
// Lane l+16     : streams floats [hw/2, hw) of row (bc0 + l)
// Per-lane partial sums (acc0, acc1) form the A fragment of
// V_WMMA_F32_16X16X4_F32; B = all-ones, so D[m][*] = full row-m sum.
__global__ __launch_bounds__(256) void gem_pool_wmma_kernel(
    const float* __restrict__ x,
    const float* __restrict__ params,
    float* __restrict__ out,
    int n_bc, int C, int hw)
{
    const int lane   = threadIdx.x & 31;
    const int gwave  = blockIdx.x * (blockDim.x >> 5) + (threadIdx.x >> 5);
    const int bc0    = gwave << 4;                 // first of 16 rows for this wave
    if (bc0 >= n_bc) return;                       // wave-uniform; EXEC stays all-1s

    const int   r    = lane & 15;                  // row-in-tile
    const int   row  = bc0 + r;                    // global (b,c) index
    const int   c0   = bc0 % C;                    // channel of row 0 (C % 16 == 0)
    const float p    = params[c0 + r];

    const int   half         = lane >> 4;          // which half of the row
    const int   nfloats_half = hw >> 1;            // 392
    const int   nf4          = nfloats_half >> 2;  // 98 float4 loads per lane
    const v4f*  bp = (const v4f*)(x + (size_t)row * (size_t)hw
                                    + (size_t)half * (size_t)nfloats_half);

    float acc0 = 0.0f, acc1 = 0.0f;

    if (__all(p == 3.0f)) {
        // fast path: cube instead of pow (reference uses p = 3.0 everywhere)
        #pragma unroll 4
        for (int k = 0; k < nf4; ++k) {
            v4f v = __builtin_nontemporal_load(bp + k);   // streamed once: NT hint
            float m0 = fmaxf(v.x, GEM_EPS), m1 = fmaxf(v.y, GEM_EPS);
            float m2 = fmaxf(v.z, GEM_EPS), m3 = fmaxf(v.w, GEM_EPS);
            acc0 += m0 * m0 * m0 + m1 * m1 * m1;
            acc1 += m2 * m2 * m2 + m3 * m3 * m3;
        }
    } else {
        // general path: m^p = exp2(p * log2(m)); m >= EPS > 0 so log2 is safe
        #pragma unroll 2
        for (int k = 0; k < nf4; ++k) {
            v4f v = __builtin_nontemporal_load(bp + k);
            float m0 = fmaxf(v.x, GEM_EPS), m1 = fmaxf(v.y, GEM_EPS);
            float m2 = fmaxf(v.z, GEM_EPS), m3 = fmaxf(v.w, GEM_EPS);
            acc0 += fast_exp2(p * fast_log2(m0)) + fast_exp2(p * fast_log2(m1));
            acc1 += fast_exp2(p * fast_log2(m2)) + fast_exp2(p * fast_log2(m3));
        }
    }

    // Cross-lane reduction in one matrix op:
    // A (16x4 f32): lane l<16 -> row l, K=0,1 ; lane l+16 -> row l, K=2,3
    // B (4x16) = all ones  =>  D[m][n] = sum_k A[m][k] = full row-m sum.
    v2f a; a.x = acc0; a.y = acc1;
    v2f b; b.x = 1.0f; b.y = 1.0f;
    v8f c = {};
    v8f d = __builtin_amdgcn_wmma_f32_16x16x4_f32(
        /*neg_a=*/false, a, /*neg_b=*/false, b,
        /*c_mod=*/(short)0, c, /*reuse_a=*/false, /*reuse_b=*/false);

    // C/D layout: VGPR i on lanes 0-15 holds D[i][n], on lanes 16-31 holds D[8+i][n].
    // Every column n is identical, so lane 0 and lane 16 each finalize 8 outputs.
    const float inv_hw = 1.0f / (float)hw;
    if (lane == 0) {
        #pragma unroll
        for (int i = 0; i < 8; ++i) {
            float pp   = params[c0 + i];
            float mean = d[i] * inv_hw;
            out[bc0 + i] = fast_exp2(fast_log2(mean) / pp);   // mean^(1/p)
        }
    } else if (lane == 16) {
        #pragma unroll
        for (int i = 0; i < 8; ++i) {
            float pp   = params[c0 + 8 + i];
            float mean = d[i] * inv_hw;
            out[bc0 + 8 + i] = fast_exp2(fast_log2(mean) / pp);
        }
    }
}

extern "C" void kernel_launch(void* const* d_in, const int* in_sizes, int n_in,
                              void* d_out, int out_size, void* d_ws, size_t ws_size,
                              hipStream_t stream) {
    const float* x      = (const float*)d_in[0];   // (32, 2048, 28, 28) f32
    const float* params = (const float*)d_in[1];   // (2048,) f32
    float*       out    = (float*)d_out;           // (32, 2048, 1, 1) f32 flat

    const int n_bc = out_size;                 // 32 * 2048 = 65536
    const int C    = in_sizes[1];              // 2048
    const int hw   = in_sizes[0] / out_size;   // 784

    const int waves   = n_bc / 16;             // 16 rows per wave
    const int threads = 256;                   // 8 waves per block
    const int wpb     = threads / 32;
    const int blocks  = (waves + wpb - 1) / wpb;

    gem_pool_wmma_kernel<<<blocks, threads, 0, stream>>>(x, params, out, n_bc, C, hw);
}